// TemporalGCN_38568806318196
// MI455X (gfx1250) — compile-verified
//
#include <hip/hip_runtime.h>
#include <math.h>

typedef __attribute__((ext_vector_type(2))) float v2f;
typedef __attribute__((ext_vector_type(8))) float v8f;

#define N_NODES 20000
#define F_IN    8
#define P_PER   12
#define O_HID   32
#define E_EDGES 320000

// workspace layout (float offsets)
#define WS_DINV   0
#define WS_AGG    (WS_DINV + N_NODES)
#define WS_AZ     (WS_AGG + N_NODES * P_PER * F_IN)
#define WS_AH     (WS_AZ + F_IN * O_HID)
#define WS_CZ     (WS_AH + F_IN * O_HID)
#define WS_CH     (WS_CZ + O_HID)
#define WS_PROBS  (WS_CH + O_HID)

#define LOG2E 1.442695040888963f

// branch-free fast transcendentals: single v_exp_f32 / v_rcp_f32 each,
// no EXEC divergence inside the WMMA loop (args are O(0.1) -> no overflow)
__device__ __forceinline__ float fast_sigmoid(float x) {
    return __builtin_amdgcn_rcpf(1.0f + __builtin_amdgcn_exp2f(-LOG2E * x));
}
__device__ __forceinline__ float fast_tanh(float x) {
    return 1.0f - 2.0f * __builtin_amdgcn_rcpf(1.0f + __builtin_amdgcn_exp2f((2.0f * LOG2E) * x));
}

// ---------------------------------------------------------------- degree ----
__global__ void k_deg_init(float* deg) {
    int i = blockIdx.x * blockDim.x + threadIdx.x;
    if (i < N_NODES) deg[i] = 1.0f;           // self-loop contributes 1
}

__global__ void k_deg_edges(const int* __restrict__ ei, float* deg) {
    int e = blockIdx.x * blockDim.x + threadIdx.x;
    if (e < E_EDGES) atomicAdd(&deg[ei[E_EDGES + e]], 1.0f);  // dst row
}

__global__ void k_dinv(float* deg) {
    int i = blockIdx.x * blockDim.x + threadIdx.x;
    if (i < N_NODES) {
        float d = deg[i];
        deg[i] = (d > 0.0f) ? rsqrtf(d) : 0.0f;   // in place: deg -> d^-1/2
    }
}

// ------------------------------------------------------------ aggregation ---
// agg[i][p][f] = sum over incoming edges (+ self loop) of norm * x[src][f][p]
// x layout: [N][F][P] (p fastest). agg layout: [N][P][F] (f fastest, WMMA-A friendly)
__global__ void k_agg_self(const float* __restrict__ x,
                           const float* __restrict__ dinv,
                           float* __restrict__ agg) {
    int idx = blockIdx.x * blockDim.x + threadIdx.x;
    if (idx < N_NODES * 96) {
        int i = idx / 96, r = idx % 96;
        int p = r >> 3, f = r & 7;
        float w = dinv[i] * dinv[i];          // self-loop norm
        agg[i * 96 + p * 8 + f] = w * x[i * 96 + f * 12 + p];
    }
}

__global__ void k_agg_edges(const float* __restrict__ x,
                            const int* __restrict__ ei,
                            const float* __restrict__ dinv,
                            float* __restrict__ agg) {
    int idx = blockIdx.x * blockDim.x + threadIdx.x;
    if (idx < E_EDGES * 96) {
        int e = idx / 96, r = idx % 96;
        int p = r >> 3, f = r & 7;
        int s = ei[e], d = ei[E_EDGES + e];
        float w = dinv[s] * dinv[d];
        atomicAdd(&agg[d * 96 + p * 8 + f], w * x[s * 96 + f * 12 + p]);
    }
}

// -------------------------------------------- fold weight chains + softmax --
// Az[f][j] = sum_k Wz[f][k] * lz_w[j][k]   (k < 32, lz_w row stride = 64)
// cz[j]    = sum_k bz[k]   * lz_w[j][k] + lz_b[j]     (same for h-gate)
__global__ void k_pre(const float* __restrict__ Wz, const float* __restrict__ bz,
                      const float* __restrict__ Wh, const float* __restrict__ bh,
                      const float* __restrict__ lzw, const float* __restrict__ lzb,
                      const float* __restrict__ lhw, const float* __restrict__ lhb,
                      const float* __restrict__ att,
                      float* Az, float* Ah, float* cz, float* ch, float* probs) {
    int t = threadIdx.x;
    {
        int f = t >> 5, j = t & 31;
        float az = 0.0f, ah = 0.0f;
        for (int k = 0; k < 32; ++k) {
            az += Wz[f * 32 + k] * lzw[j * 64 + k];
            ah += Wh[f * 32 + k] * lhw[j * 64 + k];
        }
        Az[f * 32 + j] = az;
        Ah[f * 32 + j] = ah;
    }
    if (t < 32) {
        float sz = lzb[t], sh = lhb[t];
        for (int k = 0; k < 32; ++k) {
            sz += bz[k] * lzw[t * 64 + k];
            sh += bh[k] * lhw[t * 64 + k];
        }
        cz[t] = sz;
        ch[t] = sh;
    }
    if (t == 0) {
        float m = att[0];
        for (int p = 1; p < P_PER; ++p) m = fmaxf(m, att[p]);
        float ex[P_PER]; float s = 0.0f;
        for (int p = 0; p < P_PER; ++p) { ex[p] = __expf(att[p] - m); s += ex[p]; }
        float inv = 1.0f / s;
        for (int p = 0; p < P_PER; ++p) probs[p] = ex[p] * inv;
    }
}

// --------------------------------------------------------- fused WMMA core --
// One wave (32 lanes) owns a 16-node tile; 8 waves per 256-thread block.
// Per period p: D = A(16x8 agg) x B(8x32 folded gate weights), fp32 WMMA k=4.
// Sigmoid/tanh/(1-Z)*Ht fused in C-layout, attention-weighted into Hacc.
// Finally Hacc goes C-layout -> LDS -> A-layout for the 32->12 projection WMMAs.
// Period loop kept ROLLED to stay under 256 VGPRs (>=4 waves/SIMD).
__global__ __launch_bounds__(256, 4) void k_fused(
        const float* __restrict__ agg,
        const float* __restrict__ Az, const float* __restrict__ Ah,
        const float* __restrict__ cz, const float* __restrict__ ch,
        const float* __restrict__ probs,
        const float* __restrict__ out_w, const float* __restrict__ out_b,
        float* __restrict__ out) {
    __shared__ float lds[8][16 * 33];      // per-wave 16x32 tile, padded stride

    const int lane = threadIdx.x & 31;
    const int wave = threadIdx.x >> 5;
    const int half = lane >> 4;            // 0: K pair {0,1}, 1: K pair {2,3}
    const int l16  = lane & 15;
    const int tileBase = (blockIdx.x * 8 + wave) * 16;

    // --- B operands (resident in regs): Bg[kstep s][ntile j], per-lane v2f ---
    v2f Bz[2][2], Bh[2][2];
#pragma unroll
    for (int s = 0; s < 2; ++s)
#pragma unroll
        for (int j = 0; j < 2; ++j) {
            int k0  = 4 * s + 2 * half;
            int col = 16 * j + l16;
            Bz[s][j][0] = Az[(k0)     * 32 + col];
            Bz[s][j][1] = Az[(k0 + 1) * 32 + col];
            Bh[s][j][0] = Ah[(k0)     * 32 + col];
            Bh[s][j][1] = Ah[(k0 + 1) * 32 + col];
        }
    float czb[2], chb[2];
    czb[0] = cz[l16]; czb[1] = cz[16 + l16];
    chb[0] = ch[l16]; chb[1] = ch[16 + l16];

    // clamped row pointer (no EXEC divergence before WMMAs)
    int node = tileBase + l16;
    if (node > N_NODES - 1) node = N_NODES - 1;
    const float* arow = agg + node * 96 + 2 * half;

    v8f hacc[2] = {};                      // attention-weighted Hp, C-layout
#pragma unroll 1
    for (int p = 0; p < P_PER; ++p) {
        float prp = probs[p];                       // uniform -> s_load
        v2f a0 = *(const v2f*)(arow + p * 8);       // K-step 0 (f 0..3)
        v2f a1 = *(const v2f*)(arow + p * 8 + 4);   // K-step 1 (f 4..7)
#pragma unroll
        for (int j = 0; j < 2; ++j) {
            v8f az8 = {};
            az8 = __builtin_amdgcn_wmma_f32_16x16x4_f32(false, a0, false, Bz[0][j], (short)0, az8, false, false);
            az8 = __builtin_amdgcn_wmma_f32_16x16x4_f32(false, a1, false, Bz[1][j], (short)0, az8, false, false);
            v8f ah8 = {};
            ah8 = __builtin_amdgcn_wmma_f32_16x16x4_f32(false, a0, false, Bh[0][j], (short)0, ah8, false, false);
            ah8 = __builtin_amdgcn_wmma_f32_16x16x4_f32(false, a1, false, Bh[1][j], (short)0, ah8, false, false);
#pragma unroll
            for (int v = 0; v < 8; ++v) {
                float z  = fast_sigmoid(az8[v] + czb[j]);
                float ht = fast_tanh(ah8[v] + chb[j]);
                hacc[j][v] += prp * (1.0f - z) * ht;
            }
        }
    }

    // relu(Hacc) -> LDS: element (m = v + 8*half, o = l16 + 16*j)
#pragma unroll
    for (int j = 0; j < 2; ++j)
#pragma unroll
        for (int v = 0; v < 8; ++v)
            lds[wave][(v + 8 * half) * 33 + (l16 + 16 * j)] = fmaxf(hacc[j][v], 0.0f);
    __syncthreads();

    // final projection: A = relu(Hacc) 16x32 from LDS, B = out_w^T (32 x 12)
    int tcol = (l16 < P_PER) ? l16 : (P_PER - 1);   // clamp N-index for loads
    v8f acc = {};
#pragma unroll
    for (int s = 0; s < 8; ++s) {
        int k0 = 4 * s + 2 * half;
        v2f a;
        a[0] = lds[wave][l16 * 33 + k0];
        a[1] = lds[wave][l16 * 33 + k0 + 1];
        v2f b = *(const v2f*)(out_w + tcol * 32 + k0);   // row-major [12][32]
        acc = __builtin_amdgcn_wmma_f32_16x16x4_f32(false, a, false, b, (short)0, acc, false, false);
    }
    float ob = out_b[tcol];
#pragma unroll
    for (int v = 0; v < 8; ++v) {
        int m  = v + 8 * half;            // C-layout row
        int nd = tileBase + m;
        if (nd < N_NODES && l16 < P_PER)
            out[nd * P_PER + l16] = acc[v] + ob;
    }
}

// ------------------------------------------------------------------ launch --
extern "C" void kernel_launch(void* const* d_in, const int* in_sizes, int n_in,
                              void* d_out, int out_size, void* d_ws, size_t ws_size,
                              hipStream_t stream) {
    const float* x   = (const float*)d_in[0];
    const int*   ei  = (const int*)  d_in[1];
    const float* Wz  = (const float*)d_in[2];
    const float* bz  = (const float*)d_in[3];
    // d_in[4], d_in[5] (Wr, br) and d_in[10], d_in[11] (lr_*) are provably
    // dead: they only ever multiply H which stays zero in the reference.
    const float* Wh  = (const float*)d_in[6];
    const float* bh  = (const float*)d_in[7];
    const float* lzw = (const float*)d_in[8];
    const float* lzb = (const float*)d_in[9];
    const float* lhw = (const float*)d_in[12];
    const float* lhb = (const float*)d_in[13];
    const float* att = (const float*)d_in[14];
    const float* ow  = (const float*)d_in[15];
    const float* ob  = (const float*)d_in[16];

    float* ws    = (float*)d_ws;
    float* dinv  = ws + WS_DINV;
    float* agg   = ws + WS_AGG;
    float* Az    = ws + WS_AZ;
    float* Ah    = ws + WS_AH;
    float* czv   = ws + WS_CZ;
    float* chv   = ws + WS_CH;
    float* probs = ws + WS_PROBS;

    k_pre<<<1, 256, 0, stream>>>(Wz, bz, Wh, bh, lzw, lzb, lhw, lhb, att,
                                 Az, Ah, czv, chv, probs);

    k_deg_init<<<(N_NODES + 255) / 256, 256, 0, stream>>>(dinv);
    k_deg_edges<<<(E_EDGES + 255) / 256, 256, 0, stream>>>(ei, dinv);
    k_dinv<<<(N_NODES + 255) / 256, 256, 0, stream>>>(dinv);

    k_agg_self<<<(N_NODES * 96 + 255) / 256, 256, 0, stream>>>(x, dinv, agg);
    k_agg_edges<<<(E_EDGES * 96 + 255) / 256, 256, 0, stream>>>(x, ei, dinv, agg);

    // 8 waves/block, 16 nodes/wave -> 128 nodes/block
    k_fused<<<(N_NODES + 127) / 128, 256, 0, stream>>>(
        agg, Az, Ah, czv, chv, probs, ow, ob, (float*)d_out);
}